// RoGERModel_34668976013908
// MI455X (gfx1250) — compile-verified
//
#include <hip/hip_runtime.h>
#include <hip/hip_bf16.h>

#define NUM_USERS 60000
#define NUM_ITEMS 30000
#define NNODES    90000
#define EK        64
#define EHALF     1000000
#define NEDGES    2000000
#define NLAYERS   3
#define NBATCH    100000
#define LMIX      0.5f
#define EPSV      1e-8f

typedef float v2f __attribute__((ext_vector_type(2)));
typedef float v8f __attribute__((ext_vector_type(8)));

// ---------------------------------------------------------------------------
// WMMA f32 GEMM:  C[M x 64] = A[M x 64] @ W[64 x 64] (+ bias)
// One wave per 16x64 output tile (4 accumulators): A is loaded exactly once
// per tile, and each A fragment feeds 4 independent back-to-back WMMAs (one
// per 16-col n-tile), hiding XDL latency and minimizing HBM traffic on the
// dominant A stream.
// A-frag layout (32-bit A 16x4): lane L holds row (L&15), cols {k0+2*(L>>4), +1}
//   -> contiguous float2 load from row-major A.
// B-frag layout (row-striped): lane L holds col n0+(L&15), rows k0+2*(L>>4), +1.
// C/D layout: VGPR v, lane L -> row m0 + v + 8*(L>>4), col n0 + (L&15).
// ---------------------------------------------------------------------------
__global__ __launch_bounds__(32) void gemm64_wmma(const float* __restrict__ A,
                                                  const float* __restrict__ W,
                                                  const float* __restrict__ bias,
                                                  float* __restrict__ C) {
    const int m0   = blockIdx.x * 16;
    const int lane = threadIdx.x;
    const int half = lane >> 4;          // 0: K even pair, 1: K odd pair
    const int l16  = lane & 15;

    const float* __restrict__ arow = A + (long)(m0 + l16) * EK;
    const float* __restrict__ wcol = W + l16;   // + n-tile*16 + k*64

    v8f acc0 = {}, acc1 = {}, acc2 = {}, acc3 = {};
#pragma unroll
    for (int k0 = 0; k0 < EK; k0 += 4) {
        const int k = k0 + 2 * half;
        v2f a = *(const v2f*)(arow + k);                  // A[row][k], A[row][k+1]
        const float* __restrict__ wk = wcol + (long)k * EK;
        v2f b0, b1, b2, b3;
        b0.x = wk[0];        b0.y = wk[EK];               // n-tile 0
        b1.x = wk[16];       b1.y = wk[EK + 16];          // n-tile 1
        b2.x = wk[32];       b2.y = wk[EK + 32];          // n-tile 2
        b3.x = wk[48];       b3.y = wk[EK + 48];          // n-tile 3
        acc0 = __builtin_amdgcn_wmma_f32_16x16x4_f32(false, a, false, b0,
                                                     (short)0, acc0, false, false);
        acc1 = __builtin_amdgcn_wmma_f32_16x16x4_f32(false, a, false, b1,
                                                     (short)0, acc1, false, false);
        acc2 = __builtin_amdgcn_wmma_f32_16x16x4_f32(false, a, false, b2,
                                                     (short)0, acc2, false, false);
        acc3 = __builtin_amdgcn_wmma_f32_16x16x4_f32(false, a, false, b3,
                                                     (short)0, acc3, false, false);
    }

    float bv0 = 0.f, bv1 = 0.f, bv2 = 0.f, bv3 = 0.f;
    if (bias) {
        bv0 = bias[l16];
        bv1 = bias[16 + l16];
        bv2 = bias[32 + l16];
        bv3 = bias[48 + l16];
    }
#pragma unroll
    for (int v = 0; v < 8; ++v) {
        float* __restrict__ crow = C + (long)(m0 + v + 8 * half) * EK + l16;
        crow[0]  = acc0[v] + bv0;
        crow[16] = acc1[v] + bv1;
        crow[32] = acc2[v] + bv2;
        crow[48] = acc3[v] + bv3;
    }
}

// ---------------------------------------------------------------------------
__global__ __launch_bounds__(256) void zero_f32(float* __restrict__ p, int n) {
    int i = blockIdx.x * 256 + threadIdx.x;
    if (i < n) p[i] = 0.0f;
}

__global__ __launch_bounds__(256) void concat_x0(const float* __restrict__ Gu,
                                                 const float* __restrict__ Gi,
                                                 float* __restrict__ x) {
    int i = blockIdx.x * 256 + threadIdx.x;
    const int nu = NUM_USERS * EK, tot = NNODES * EK;
    if (i < nu)       x[i] = Gu[i];
    else if (i < tot) x[i] = Gi[i - nu];
}

// per-edge relu(cosine(a,b)); a = x[row]*proj, b = x[row-NU]*proj
__global__ __launch_bounds__(256) void edge_sim(const float* __restrict__ x,
                                                const float* __restrict__ proj,
                                                const int* __restrict__ edge_row,
                                                float* __restrict__ sim) {
    int e = blockIdx.x * 256 + threadIdx.x;
    if (e >= EHALF) return;
    const int r = edge_row[e];
    const float4* pr = (const float4*)(proj + (long)e * EK);
    const float4* rn = (const float4*)(x + (long)r * EK);
    const float4* cn = (const float4*)(x + (long)(r - NUM_USERS) * EK);
    float saa = 0.f, sbb = 0.f, sab = 0.f;
#pragma unroll
    for (int d = 0; d < EK / 4; ++d) {
        float4 p = pr[d], rv = rn[d], cv = cn[d];
        float ax = rv.x * p.x, ay = rv.y * p.y, az = rv.z * p.z, aw = rv.w * p.w;
        float bx = cv.x * p.x, by = cv.y * p.y, bz = cv.z * p.z, bw = cv.w * p.w;
        saa += ax * ax + ay * ay + az * az + aw * aw;
        sbb += bx * bx + by * by + bz * bz + bw * bw;
        sab += ax * bx + ay * by + az * bz + aw * bw;
    }
    float na = fmaxf(sqrtf(saa), EPSV);
    float nb = fmaxf(sqrtf(sbb), EPSV);
    sim[e] = fmaxf(sab / (na * nb), 0.0f);
}

__global__ __launch_bounds__(256) void vals_deg(const float* __restrict__ L0,
                                                const float* __restrict__ sim,
                                                const int* __restrict__ edge_row,
                                                float* __restrict__ vals,
                                                float* __restrict__ deg) {
    int e = blockIdx.x * 256 + threadIdx.x;
    if (e >= NEDGES) return;
    float s = sim[e < EHALF ? e : e - EHALF];
    float v = LMIX * L0[e] + (1.0f - LMIX) * s;
    vals[e] = v;
    __hip_atomic_fetch_add(&deg[edge_row[e]], v, __ATOMIC_RELAXED,
                           __HIP_MEMORY_SCOPE_AGENT);
}

__global__ __launch_bounds__(256) void dinv_k(const float* __restrict__ deg,
                                              float* __restrict__ dinv) {
    int i = blockIdx.x * 256 + threadIdx.x;
    if (i >= NNODES) return;
    float d = deg[i];
    dinv[i] = d > 0.0f ? rsqrtf(d) : 0.0f;
}

__global__ __launch_bounds__(256) void ew_k(const float* __restrict__ dinv,
                                            const float* __restrict__ vals,
                                            const int* __restrict__ edge_row,
                                            const int* __restrict__ edge_col,
                                            float* __restrict__ ew) {
    int e = blockIdx.x * 256 + threadIdx.x;
    if (e >= NEDGES) return;
    ew[e] = dinv[edge_row[e]] * vals[e] * dinv[edge_col[e]];
}

// (edge, dim) threads: 64 lanes/edge -> coalesced gather of h[col][:],
// coalesced atomic scatter-add burst into agg[row][:] (L2-resident, 23 MB).
__global__ __launch_bounds__(256) void edge_agg(const float* __restrict__ h,
                                                const float* __restrict__ ew,
                                                const int* __restrict__ edge_row,
                                                const int* __restrict__ edge_col,
                                                float* __restrict__ agg) {
    int tid = blockIdx.x * 256 + threadIdx.x;
    if (tid >= NEDGES * EK) return;
    int e = tid >> 6, d = tid & 63;
    int r = edge_row[e], c = edge_col[e];
    __hip_atomic_fetch_add(&agg[(long)r * EK + d], ew[e] * h[(long)c * EK + d],
                           __ATOMIC_RELAXED, __HIP_MEMORY_SCOPE_AGENT);
}

__global__ __launch_bounds__(256) void relu_bias(float* __restrict__ x,
                                                 const float* __restrict__ b) {
    int i = blockIdx.x * 256 + threadIdx.x;
    if (i >= NNODES * EK) return;
    x[i] = fmaxf(x[i] + b[i & 63], 0.0f);
}

__global__ __launch_bounds__(256) void final_score(const float* __restrict__ x,
                                                   const float* __restrict__ Bu,
                                                   const float* __restrict__ Bi,
                                                   const float* __restrict__ Mu,
                                                   const int* __restrict__ user_idx,
                                                   const int* __restrict__ item_idx,
                                                   float* __restrict__ out) {
    int b = blockIdx.x * 256 + threadIdx.x;
    if (b >= NBATCH) return;
    int u = user_idx[b], it = item_idx[b];
    const float4* gu = (const float4*)(x + (long)u * EK);
    const float4* gi = (const float4*)(x + (long)(NUM_USERS + it) * EK);
    float s = 0.f;
#pragma unroll
    for (int d = 0; d < EK / 4; ++d) {
        float4 a = gu[d], c = gi[d];
        s += a.x * c.x + a.y * c.y + a.z * c.z + a.w * c.w;
    }
    out[b] = s + Bu[u] + Bi[it] + Mu[0];
}

// ---------------------------------------------------------------------------
extern "C" void kernel_launch(void* const* d_in, const int* in_sizes, int n_in,
                              void* d_out, int out_size, void* d_ws, size_t ws_size,
                              hipStream_t stream) {
    const float* Gu       = (const float*)d_in[0];
    const float* Gi       = (const float*)d_in[1];
    const float* Bu       = (const float*)d_in[2];
    const float* Bi       = (const float*)d_in[3];
    const float* Mu       = (const float*)d_in[4];
    const float* proj_W   = (const float*)d_in[5];
    const float* proj_b   = (const float*)d_in[6];
    const float* gcn_W    = (const float*)d_in[7];
    const float* gcn_b    = (const float*)d_in[8];
    const float* edge_feat= (const float*)d_in[9];
    const float* L0       = (const float*)d_in[10];
    const int*   edge_row = (const int*)d_in[11];
    const int*   edge_col = (const int*)d_in[12];
    const int*   user_idx = (const int*)d_in[13];
    const int*   item_idx = (const int*)d_in[14];
    float*       out      = (float*)d_out;

    // workspace carving (256B aligned); total ~346 MB
    char* w = (char*)d_ws;
    auto carve = [&](size_t bytes) {
        char* p = w;
        w += (bytes + 255) & ~(size_t)255;
        return (float*)p;
    };
    float* x0   = carve(sizeof(float) * (size_t)NNODES * EK);
    float* x1   = carve(sizeof(float) * (size_t)NNODES * EK);
    float* proj = carve(sizeof(float) * (size_t)EHALF * EK);
    float* sim  = carve(sizeof(float) * (size_t)EHALF);
    float* vals = carve(sizeof(float) * (size_t)NEDGES);
    float* ew   = carve(sizeof(float) * (size_t)NEDGES);
    float* deg  = carve(sizeof(float) * (size_t)NNODES);
    float* dinv = carve(sizeof(float) * (size_t)NNODES);
    float* h    = carve(sizeof(float) * (size_t)NNODES * EK);

    const int NXE = NNODES * EK;         // 5,760,000
    concat_x0<<<(NXE + 255) / 256, 256, 0, stream>>>(Gu, Gi, x0);

    // proj = edge_feat @ proj_W + proj_b   (WMMA, 1M rows -> 62500 wave-tiles)
    gemm64_wmma<<<EHALF / 16, 32, 0, stream>>>(edge_feat, proj_W, proj_b, proj);

    float* cur = x0;
    float* nxt = x1;
    for (int l = 0; l < NLAYERS; ++l) {
        edge_sim<<<(EHALF + 255) / 256, 256, 0, stream>>>(cur, proj, edge_row, sim);

        zero_f32<<<(NNODES + 255) / 256, 256, 0, stream>>>(deg, NNODES);
        vals_deg<<<(NEDGES + 255) / 256, 256, 0, stream>>>(L0, sim, edge_row, vals, deg);
        dinv_k<<<(NNODES + 255) / 256, 256, 0, stream>>>(deg, dinv);
        ew_k<<<(NEDGES + 255) / 256, 256, 0, stream>>>(dinv, vals, edge_row, edge_col, ew);

        // h = cur @ gcn_W[l]   (WMMA, 90000 rows -> 5625 wave-tiles)
        gemm64_wmma<<<NNODES / 16, 32, 0, stream>>>(
            cur, gcn_W + (size_t)l * EK * EK, nullptr, h);

        zero_f32<<<(NXE + 255) / 256, 256, 0, stream>>>(nxt, NXE);
        edge_agg<<<(NEDGES * EK) / 256, 256, 0, stream>>>(h, ew, edge_row, edge_col, nxt);
        relu_bias<<<(NXE + 255) / 256, 256, 0, stream>>>(nxt, gcn_b + (size_t)l * EK);

        float* t = cur; cur = nxt; nxt = t;
    }

    final_score<<<(NBATCH + 255) / 256, 256, 0, stream>>>(
        cur, Bu, Bi, Mu, user_idx, item_idx, out);
    (void)in_sizes; (void)n_in; (void)out_size; (void)ws_size;
}